// GCN3_44023414784199
// MI455X (gfx1250) — compile-verified
//
#include <hip/hip_runtime.h>

// ---------------------------------------------------------------------------
// GCN3 on MI455X (gfx1250, wave32).
//
// Cost model: scatter phase moves ~1.2 GB irregularly (HBM roofline ~50us @
// 23.3 TB/s); agg[] is 12.8 MB so the 153M f32 atomics land in the 192 MB L2.
// The dense 32x32 layer GEMMs are ~0.6 GFLOP total -> done with
// V_WMMA_F32_16X16X4_F32 (f32 in/out keeps the segment-sum precision).
// ---------------------------------------------------------------------------

typedef __attribute__((ext_vector_type(2))) float v2f;
typedef __attribute__((ext_vector_type(8))) float v8f;

#define GCN_N_NODES 100000
#define GCN_N_EDGES 1600000

__global__ void zero_f32(float* __restrict__ p, int n) {
  int i = blockIdx.x * blockDim.x + threadIdx.x;
  if (i < n) p[i] = 0.0f;
}

// One wave (32 lanes) per edge: lane f handles feature f.
// Row load of h[src] is one coalesced 128B burst; 32 lane-parallel
// global_atomic_add_f32 into agg[dst] (resolves in L2: agg is 12.8MB).
__global__ void scatter_add_32(const float* __restrict__ h,
                               const int* __restrict__ src,
                               const int* __restrict__ dst,
                               float* __restrict__ agg,
                               int n_edges) {
  int e = (int)((blockIdx.x * blockDim.x + threadIdx.x) >> 5);  // wave-uniform
  int f = threadIdx.x & 31;
  if (e >= n_edges) return;
  int s = src[e];
  int d = dst[e];
  float v = h[(size_t)s * 32 + f];
  atomicAdd(&agg[(size_t)d * 32 + f], v);
}

// out[n, NOUT] = act(A[n,32] @ W[32,NOUT] + bias), via V_WMMA_F32_16X16X4_F32.
// One wave owns a 16-row tile; NT = NOUT/16 accumulators; K=32 in 8 steps of 4.
//
// A frag (16x4 f32, ISA 7.12.2): lane L (L<16) holds {A[L][k0], A[L][k0+1]},
// lanes 16-31 hold k0+2, k0+3 for row L-16 -> per-lane v2f, 8B-aligned b64 load.
// B frag (4x16): lane L holds column (L&15) rows {kb, kb+1} (kb += 2 for upper
// half-wave). C/D (16x16 f32): VGPR i = {M=i lanes 0-15, M=8+i lanes 16-31}.
template <int NOUT, bool RELU>
__global__ void gemm_bias_act(const float* __restrict__ A,     // [n, 32]
                              const float* __restrict__ W,     // [32, NOUT]
                              const float* __restrict__ bias,  // [NOUT]
                              float* __restrict__ out,         // [n, NOUT]
                              int n) {
  constexpr int NT = NOUT / 16;
  int wave = (int)((blockIdx.x * blockDim.x + threadIdx.x) >> 5);
  int lane = threadIdx.x & 31;
  int row0 = wave * 16;
  if (row0 >= n) return;  // wave-uniform: EXEC stays all-1s for WMMA

  int m = lane & 15;            // row (A) / column (B,C,D) within the tile
  int khalf = (lane >> 4) * 2;  // +0 for lanes 0-15, +2 for lanes 16-31

  v8f acc[NT];
#pragma unroll
  for (int t = 0; t < NT; ++t)
#pragma unroll
    for (int i = 0; i < 8; ++i) acc[t][i] = 0.0f;

  const float* arow = A + (size_t)(row0 + m) * 32;
#pragma unroll
  for (int kk = 0; kk < 8; ++kk) {
    int kb = kk * 4 + khalf;                 // even -> 8B aligned
    v2f a = *(const v2f*)(arow + kb);        // b64 load of A fragment
#pragma unroll
    for (int t = 0; t < NT; ++t) {
      int col = t * 16 + m;
      v2f b;
      b.x = W[(size_t)kb * NOUT + col];
      b.y = W[(size_t)(kb + 1) * NOUT + col];
      // (neg_a, A, neg_b, B, c_mod, C, reuse_a, reuse_b)
      acc[t] = __builtin_amdgcn_wmma_f32_16x16x4_f32(
          false, a, false, b, (short)0, acc[t], false, false);
    }
  }

  int mbase = row0 + ((lane >> 4) << 3);  // lanes 16-31 hold rows +8..+15
#pragma unroll
  for (int i = 0; i < 8; ++i) {
    int r = mbase + i;
#pragma unroll
    for (int t = 0; t < NT; ++t) {
      int col = t * 16 + m;
      float v = acc[t][i] + bias[col];
      if (RELU) v = (v >= 0.0f) ? v : 0.1f * v;
      out[(size_t)r * NOUT + col] = v;
    }
  }
}

extern "C" void kernel_launch(void* const* d_in, const int* in_sizes, int n_in,
                              void* d_out, int out_size, void* d_ws, size_t ws_size,
                              hipStream_t stream) {
  const float* x  = (const float*)d_in[0];
  const int*   src = (const int*)d_in[1];
  const int*   dst = (const int*)d_in[2];
  const float* W1 = (const float*)d_in[3];
  const float* b1 = (const float*)d_in[4];
  const float* W2 = (const float*)d_in[5];
  const float* b2 = (const float*)d_in[6];
  const float* W3 = (const float*)d_in[7];
  const float* b3 = (const float*)d_in[8];
  float* out = (float*)d_out;

  // Workspace: agg[N,32] and h[N,32], ping-ponged (25.6 MB total).
  float* agg = (float*)d_ws;
  float* h   = agg + (size_t)GCN_N_NODES * 32;

  const int NF = GCN_N_NODES * 32;                    // 3.2M floats
  dim3 blk(256);
  dim3 zgrid((NF + 255) / 256);                       // zero kernel
  dim3 sgrid((GCN_N_EDGES * 32 + 255) / 256);         // 51.2M threads, wave/edge
  int tiles = (GCN_N_NODES + 15) / 16;                // 6250 (exact)
  dim3 ggrid((tiles + 7) / 8);                        // 8 waves / 256-thread block

  // Layer 1: agg = segsum(x); h = lrelu(agg @ W1 + b1)
  zero_f32<<<zgrid, blk, 0, stream>>>(agg, NF);
  scatter_add_32<<<sgrid, blk, 0, stream>>>(x, src, dst, agg, GCN_N_EDGES);
  gemm_bias_act<32, true><<<ggrid, blk, 0, stream>>>(agg, W1, b1, h, GCN_N_NODES);

  // Layer 2: agg = segsum(h); h = lrelu(agg @ W2 + b2)   (h safe: read by scatter first)
  zero_f32<<<zgrid, blk, 0, stream>>>(agg, NF);
  scatter_add_32<<<sgrid, blk, 0, stream>>>(h, src, dst, agg, GCN_N_EDGES);
  gemm_bias_act<32, true><<<ggrid, blk, 0, stream>>>(agg, W2, b2, h, GCN_N_NODES);

  // Layer 3: agg = segsum(h); out = agg @ W3 + b3  (OUT_F = 16, no activation)
  zero_f32<<<zgrid, blk, 0, stream>>>(agg, NF);
  scatter_add_32<<<sgrid, blk, 0, stream>>>(h, src, dst, agg, GCN_N_EDGES);
  gemm_bias_act<16, false><<<ggrid, blk, 0, stream>>>(agg, W3, b3, out, GCN_N_NODES);
}